// LearnedFeatureUnification_79559974191162
// MI455X (gfx1250) — compile-verified
//
#include <hip/hip_runtime.h>
#include <hip/hip_bf16.h>
#include <stdint.h>

typedef __attribute__((ext_vector_type(2))) float v2f;
typedef __attribute__((ext_vector_type(8))) float v8f;
typedef __attribute__((ext_vector_type(4))) unsigned int u32x4;
typedef __attribute__((ext_vector_type(8))) int i32x8;
typedef __attribute__((ext_vector_type(4))) int i32x4;

#define C_CH 64
#define N_O  32
#define H_DIM 64
#define W_DIM 64
#define PIX_PER_BLOCK 8
#define TILE_W (PIX_PER_BLOCK + 2)     /* 10 cols incl. halo   */
#define FEAT_STRIDE (3 * TILE_W)       /* 30 floats / channel  */
#define FEAT_ELEMS (C_CH * FEAT_STRIDE) /* 1920 */

#if defined(__has_builtin)
#if __has_builtin(__builtin_amdgcn_tensor_load_to_lds) && \
    __has_builtin(__builtin_amdgcn_s_wait_tensorcnt)
#define HAVE_TDM 1
#endif
#endif
#ifndef HAVE_TDM
#define HAVE_TDM 0
#endif

__global__ __launch_bounds__(256)
void lfu_fused_kernel(const float* __restrict__ feat,
                      const float* __restrict__ basis,
                      float* __restrict__ out)
{
    __shared__ float s_feat[FEAT_ELEMS + 1];       // +1: zero pad slot
    __shared__ float s_basis[N_O * 9 + 1];         // +1: zero pad slot

    const int tid  = threadIdx.x;
    const int lane = tid & 31;
    const int wave = tid >> 5;                     // 0..7 -> pixel in tile

    const int blocks_per_img = (H_DIM * W_DIM) / PIX_PER_BLOCK;   // 512
    const int b  = blockIdx.x / blocks_per_img;
    const int r0 = blockIdx.x % blocks_per_img;
    const int h  = r0 / (W_DIM / PIX_PER_BLOCK);
    const int w0 = (r0 % (W_DIM / PIX_PER_BLOCK)) * PIX_PER_BLOCK;

    // ---- stage basis + zero pad slots ----
    for (int i = tid; i < N_O * 9; i += 256) s_basis[i] = basis[i];
    if (tid == 0) {
        s_basis[N_O * 9] = 0.0f;
        s_feat[FEAT_ELEMS] = 0.0f;
    }

    // ---- stage feature tile (64ch x 3row x 10col) ----
    // Tile origin is (h-1, w0-1). TDM zero-fills right/bottom OOB via tensor
    // dims; a negative origin (left/top image edge) needs the manual path.
    const bool tdm_ok = (h > 0) && (w0 > 0);

#if HAVE_TDM
    if (tdm_ok) {
        if (tid < 32) {                            // wave 0 drives the DMA
            uint32_t lds_base = (uint32_t)(uintptr_t)(&s_feat[0]);
            uint64_t gaddr = (uint64_t)(uintptr_t)
                (feat + (((size_t)b * C_CH) * H_DIM + (h - 1)) * (size_t)W_DIM
                      + (w0 - 1));
            // dims remaining from tile origin -> OOB beyond image returns 0
            uint32_t tdim0 = (uint32_t)(W_DIM - (w0 - 1));   // x elements
            uint32_t tdim1 = (uint32_t)(H_DIM - (h - 1));    // y rows

            u32x4 g0;
            g0[0] = 1u;                                      // count=1
            g0[1] = lds_base;                                // lds_addr
            g0[2] = (uint32_t)(gaddr & 0xFFFFFFFFu);         // global_addr lo
            g0[3] = (uint32_t)((gaddr >> 32) & 0x01FFFFFFu)  // global_addr hi
                    | (2u << 30);                            // type=2 (image)

            i32x8 g1;
            g1[0] = (int)(2u << 16);                 // data_size=4B, mask=0
            g1[1] = (int)((tdim0 & 0xFFFFu) << 16);  // tensor_dim0 lo16
            g1[2] = (int)((tdim0 >> 16) | ((tdim1 & 0xFFFFu) << 16));
            g1[3] = (int)((tdim1 >> 16) | ((uint32_t)TILE_W << 16)); // tile_dim0=10
            g1[4] = (int)(3u | (64u << 16));         // tile_dim1=3, tile_dim2=64
            g1[5] = (int)W_DIM;                      // tensor_dim0_stride lo32 = 64
            g1[6] = (int)(((uint32_t)(H_DIM * W_DIM) & 0xFFFFu) << 16); // stride1 lo16
            g1[7] = (int)((uint32_t)(H_DIM * W_DIM) >> 16);  // stride1 hi32

            i32x4 g2;
            g2[0] = C_CH;                            // tensor_dim2 = 64
            g2[1] = 0;                               // tensor_dim3 (unused)
            g2[2] = H_DIM * W_DIM;                   // tensor_dim2_stride = 4096
            g2[3] = 0;                               // stride hi + tile_dim3=0

            i32x4 g3 = {0, 0, 0, 0};

#if defined(__clang_major__) && (__clang_major__ >= 23)
            i32x8 gx = {0, 0, 0, 0, 0, 0, 0, 0};
            __builtin_amdgcn_tensor_load_to_lds(g0, g1, g2, g3, gx, 0);
#else
            __builtin_amdgcn_tensor_load_to_lds(g0, g1, g2, g3, 0);
#endif
            __builtin_amdgcn_s_wait_tensorcnt(0);
        }
    } else
#endif
    {
        for (int i = tid; i < FEAT_ELEMS; i += 256) {
            int ch  = i / FEAT_STRIDE;
            int rem = i - ch * FEAT_STRIDE;
            int dy  = rem / TILE_W;
            int dxc = rem - dy * TILE_W;
            int gh  = h - 1 + dy;
            int gw  = w0 - 1 + dxc;
            float v = 0.0f;
            if ((unsigned)gh < (unsigned)H_DIM && (unsigned)gw < (unsigned)W_DIM)
                v = feat[(((size_t)b * C_CH + ch) * H_DIM + gh) * W_DIM + gw];
            s_feat[i] = v;
        }
    }
    __syncthreads();

    const int w  = w0 + wave;
    const int hi = lane >> 4;      // half of wave: rows M+8 / K taps 2,3
    const int nn = lane & 15;      // N column within tile

    // ---- B fragments: bf[ks][tc][j] = basis[o][k], k = ks*4 + 2*hi + j ----
    // pad taps (k>=9) read the zero slot via address select (no EXEC branch)
    v2f bf[3][2];
    #pragma unroll
    for (int ks = 0; ks < 3; ++ks)
        #pragma unroll
        for (int tc = 0; tc < 2; ++tc) {
            int o = tc * 16 + nn;
            #pragma unroll
            for (int j = 0; j < 2; ++j) {
                int k = ks * 4 + 2 * hi + j;
                int idx = (k < 9) ? (o * 9 + k) : (N_O * 9);
                bf[ks][tc][j] = s_basis[idx];
            }
        }

    // ---- V = A(64x12) * B(12x32) via 4x2 tiles of V_WMMA_F32_16X16X4_F32 ----
    v8f acc[4][2] = {};
    #pragma unroll
    for (int ks = 0; ks < 3; ++ks) {
        v2f af[4];
        #pragma unroll
        for (int tr = 0; tr < 4; ++tr) {
            int cc = tr * 16 + nn;             // channel = row M
            #pragma unroll
            for (int j = 0; j < 2; ++j) {
                int k = ks * 4 + 2 * hi + j;
                int dy = k / 3, dx = k - dy * 3;
                int idx = (k < 9) ? (cc * FEAT_STRIDE + dy * TILE_W + wave + dx)
                                  : FEAT_ELEMS;
                af[tr][j] = s_feat[idx];
            }
        }
        #pragma unroll
        for (int tr = 0; tr < 4; ++tr)
            #pragma unroll
            for (int tc = 0; tc < 2; ++tc)
                acc[tr][tc] = __builtin_amdgcn_wmma_f32_16x16x4_f32(
                    false, af[tr], false, bf[ks][tc],
                    (short)0, acc[tr][tc], false, false);
    }

    // ---- border normalization ----
    float dh = 3.0f - (h == 0) - (h == H_DIM - 1);
    float dw = 3.0f - (w == 0) - (w == W_DIM - 1);
    float inv_denom = 1.0f / (dh * dw);

    float v[4][2][8];
    #pragma unroll
    for (int tr = 0; tr < 4; ++tr)
        #pragma unroll
        for (int tc = 0; tc < 2; ++tc)
            #pragma unroll
            for (int r = 0; r < 8; ++r)
                v[tr][tc][r] = acc[tr][tc][r] * inv_denom;

    // ---- softmax groups: key (par = r&1, o = tc*16+nn); axis = cc>>1 ----
    // 32 members = 16 in-lane (4 tr x 4 r) + 16 in the partner lane (xor 16).
    float e[4][2][8];
    #pragma unroll
    for (int tc = 0; tc < 2; ++tc)
        #pragma unroll
        for (int par = 0; par < 2; ++par) {
            float m = -1e30f;
            #pragma unroll
            for (int tr = 0; tr < 4; ++tr)
                #pragma unroll
                for (int r = par; r < 8; r += 2)
                    m = fmaxf(m, v[tr][tc][r]);
            m = fmaxf(m, __shfl_xor(m, 16, 32));

            float s = 0.0f;
            #pragma unroll
            for (int tr = 0; tr < 4; ++tr)
                #pragma unroll
                for (int r = par; r < 8; r += 2) {
                    float t = __expf(v[tr][tc][r] - m);
                    e[tr][tc][r] = t;
                    s += t;
                }
            s += __shfl_xor(s, 16, 32);
            float rs = __builtin_amdgcn_rcpf(s);
            #pragma unroll
            for (int tr = 0; tr < 4; ++tr)
                #pragma unroll
                for (int r = par; r < 8; r += 2)
                    e[tr][tc][r] *= rs;
        }

    // ---- mean over groups: out[m], m = 8*tr + 4*hi + q ----
    #pragma unroll
    for (int tr = 0; tr < 4; ++tr) {
        #pragma unroll
        for (int q = 0; q < 4; ++q) {
            float s2 = e[tr][0][2 * q] + e[tr][0][2 * q + 1]
                     + e[tr][1][2 * q] + e[tr][1][2 * q + 1];
            #pragma unroll
            for (int d = 1; d < 16; d <<= 1)
                s2 += __shfl_xor(s2, d, 32);
            if (nn == 0) {
                int m = tr * 8 + hi * 4 + q;
                out[(((size_t)b * N_O + m) * H_DIM + h) * W_DIM + w] =
                    s2 * (1.0f / 64.0f);
            }
        }
    }
}

extern "C" void kernel_launch(void* const* d_in, const int* in_sizes, int n_in,
                              void* d_out, int out_size, void* d_ws, size_t ws_size,
                              hipStream_t stream) {
    const float* feat  = (const float*)d_in[0];   // (8,64,64,64) f32
    const float* basis = (const float*)d_in[1];   // (32,1,3,3)  f32
    float* out = (float*)d_out;                   // (8,32,64,64) f32

    const int blocks = 8 * (H_DIM * W_DIM) / PIX_PER_BLOCK;   // 4096
    lfu_fused_kernel<<<blocks, 256, 0, stream>>>(feat, basis, out);
}